// MPC_41583873360400
// MI455X (gfx1250) — compile-verified
//
#include <hip/hip_runtime.h>

// Batched LQR (MPC) for MI455X / gfx1250.
// One wave32 per batch element; fp32 WMMA (v_wmma_f32_16x16x4_f32) for all
// dense tile products; Gauss-Jordan in LDS for the 16x16 SPD control solve.

#define TT    50
#define BATCH 2048
#define NS    32
#define NC    16
#define NF    48   // NS + NC

// Only check for the device builtin on the device compilation pass; the host
// pass (x86) never has amdgcn builtins.
#if defined(__HIP_DEVICE_COMPILE__)
#if !__has_builtin(__builtin_amdgcn_wmma_f32_16x16x4_f32)
#error "gfx1250 wmma f32 16x16x4 builtin not available on device"
#endif
#endif

typedef __attribute__((ext_vector_type(2))) float v2f;
typedef __attribute__((ext_vector_type(8))) float v8f;

// LDS leading dims (odd -> conflict-free transposed fragment reads)
#define LQ 49
#define LV 33
#define LF 49
#define LW 49
#define LU 33
#define LS 33
#define LZ 33
#define LK 33

__device__ __forceinline__ v8f zero8() {
  v8f z;
#pragma unroll
  for (int r = 0; r < 8; ++r) z[r] = 0.0f;
  return z;
}

// C/D fragment: vgpr r holds element (M = r + 8*h, N = n) ; n = lane&15, h = lane>>4
__device__ __forceinline__ v8f acc_load(const float* C, int ldc, int n, int h) {
  v8f acc;
#pragma unroll
  for (int r = 0; r < 8; ++r) acc[r] = C[(r + 8 * h) * ldc + n];
  return acc;
}
__device__ __forceinline__ void acc_store(float* D, int ldd, v8f acc, int n, int h) {
#pragma unroll
  for (int r = 0; r < 8; ++r) D[(r + 8 * h) * ldd + n] = acc[r];
}

// One 16x16 output tile, K=16 inner dim, via 4x v_wmma_f32_16x16x4_f32.
// A fragment: lane holds (m = lane&15, k = 2*h + vgpr)  [ISA 16x4 f32 A layout]
// B fragment: lane holds (n = lane&15, k = 2*h + vgpr)  [mirrors 16-bit B layout]
template <bool TRANSA>
__device__ __forceinline__ v8f mma_k16(const float* __restrict__ A, int lda,
                                       const float* __restrict__ B, int ldb,
                                       v8f acc, int mn, int h) {
#pragma unroll
  for (int kk = 0; kk < 16; kk += 4) {
    const int k0 = kk + 2 * h;
    v2f a, b;
    if (TRANSA) {  // A is used transposed: A_frag(m,k) = A[k][m]
      a.x = A[k0 * lda + mn];
      a.y = A[(k0 + 1) * lda + mn];
    } else {
      a.x = A[mn * lda + k0];
      a.y = A[mn * lda + k0 + 1];
    }
    b.x = B[k0 * ldb + mn];
    b.y = B[(k0 + 1) * ldb + mn];
#if defined(__HIP_DEVICE_COMPILE__)
    acc = __builtin_amdgcn_wmma_f32_16x16x4_f32(false, a, false, b, (short)0,
                                                acc, false, false);
#endif
  }
  return acc;
}

// coalesced global -> LDS tile copy (src row-major rows x cols, dst has ld)
__device__ __forceinline__ void g2s(float* dst, int ldd, const float* __restrict__ src,
                                    int rows, int cols, int lane) {
  for (int r = 0; r < rows; ++r)
    for (int c = lane; c < cols; c += 32)
      dst[r * ldd + c] = src[r * cols + c];
}

// Factor step: given Qm (sQ, 48x48 sym) and qv (48), compute
//   K = -Quu^{-1} Qux,  k = -Quu^{-1} qu   (Gauss-Jordan, 33 RHS)
//   V = sym(Qxx + Qxu K + (Qxu K)^T + K^T Quu K),  v = qx + Qxu k + K^T(qu + Quu k)
// Writes K,k to workspace at timestep index tb.
__device__ __forceinline__ void factor_update(
    float* sQ, float* sqv, float* sE, float* sK, float* sk, float* sU, float* sS,
    float* sZ, float* sV, float* sv, float* st1,
    float* __restrict__ Kws, float* __restrict__ kws,
    size_t tb, int lane, int h) {
  const int mn = lane & 15;

  // Build augmented system E = [Quu | Qux | qu | pad] : 16 x 64
  for (int r = 0; r < 16; ++r) {
    for (int cc = lane; cc < 64; cc += 32) {
      float val = 0.0f;
      if (cc < 16)       val = sQ[(NS + r) * LQ + NS + cc];
      else if (cc < 48)  val = sQ[(NS + r) * LQ + (cc - 16)];
      else if (cc == 48) val = sqv[NS + r];
      sE[r * 64 + cc] = val;
    }
  }
  __syncthreads();

  // Gauss-Jordan (Quu SPD, diag-dominant: no pivoting)
  for (int p = 0; p < 16; ++p) {
    const float inv = 1.0f / sE[p * 64 + p];
    const float pr0 = sE[p * 64 + lane] * inv;
    const float pr1 = sE[p * 64 + 32 + lane] * inv;
    sE[p * 64 + lane] = pr0;
    sE[p * 64 + 32 + lane] = pr1;
    __syncthreads();
    for (int r = 0; r < 16; ++r) {
      if (r == p) continue;
      const float fac = sE[r * 64 + p];  // broadcast
      sE[r * 64 + lane]      -= fac * pr0;
      sE[r * 64 + 32 + lane] -= fac * pr1;
    }
    __syncthreads();
  }

  // Extract K (16x32), k (16); negate; stash to LDS + workspace
  for (int r = 0; r < 16; ++r) {
    const float kv = -sE[r * 64 + 16 + lane];
    sK[r * LK + lane] = kv;
    Kws[tb * (size_t)(NC * NS) + r * NS + lane] = kv;
  }
  if (lane < 16) {
    const float kk2 = -sE[lane * 64 + 48];
    sk[lane] = kk2;
    kws[tb * (size_t)NC + lane] = kk2;
  }
  __syncthreads();

  // U = Qxu @ K (32x32) ; Z = Quu @ K (16x32)
#pragma unroll
  for (int ti = 0; ti < 2; ++ti)
#pragma unroll
    for (int tj = 0; tj < 2; ++tj) {
      v8f acc = zero8();
      acc = mma_k16<false>(&sQ[(16 * ti) * LQ + NS], LQ, &sK[16 * tj], LK, acc, mn, h);
      acc_store(&sU[(16 * ti) * LU + 16 * tj], LU, acc, mn, h);
    }
#pragma unroll
  for (int tj = 0; tj < 2; ++tj) {
    v8f acc = zero8();
    acc = mma_k16<false>(&sQ[NS * LQ + NS], LQ, &sK[16 * tj], LK, acc, mn, h);
    acc_store(&sZ[16 * tj], LZ, acc, mn, h);
  }
  __syncthreads();

  // S = K^T @ Z (32x32)
#pragma unroll
  for (int ti = 0; ti < 2; ++ti)
#pragma unroll
    for (int tj = 0; tj < 2; ++tj) {
      v8f acc = zero8();
      acc = mma_k16<true>(&sK[16 * ti], LK, &sZ[16 * tj], LZ, acc, mn, h);
      acc_store(&sS[(16 * ti) * LS + 16 * tj], LS, acc, mn, h);
    }
  __syncthreads();

  // v_new = qx + Qxu k + K^T (qu + Quu k)
  if (lane < 16) {
    float s = 0.0f;
    for (int c2 = 0; c2 < 16; ++c2) s += sQ[(NS + lane) * LQ + NS + c2] * sk[c2];
    st1[lane] = sqv[NS + lane] + s;  // qu + Quu k
  }
  __syncthreads();
  {
    float s = sqv[lane];  // qx
    for (int r = 0; r < 16; ++r)
      s += sQ[lane * LQ + NS + r] * sk[r] + sK[r * LK + lane] * st1[r];
    sv[lane] = s;
  }

  // V_new = Qxx + U + U^T + 0.5*(S + S^T)   (Qxx already symmetric)
  for (int i = 0; i < NS; ++i) {
    const int j = lane;
    const float val = sQ[i * LQ + j] + sU[i * LU + j] + sU[j * LU + i] +
                      0.5f * (sS[i * LS + j] + sS[j * LS + i]);
    sV[i * LV + j] = val;
  }
  __syncthreads();
}

__global__ __launch_bounds__(32) void lqr_kernel(
    const float* __restrict__ x0g, const float* __restrict__ Cg,
    const float* __restrict__ cg, const float* __restrict__ Fg,
    const float* __restrict__ fg, float* __restrict__ out,
    float* __restrict__ Kws, float* __restrict__ kws) {
  __shared__ float sQ[NF * LQ];
  __shared__ float sV[NS * LV];
  __shared__ float sF[NS * LF];
  __shared__ float sW[NS * LW];
  __shared__ float sU[NS * LU];
  __shared__ float sS[NS * LS];
  __shared__ float sZ[NC * LZ];
  __shared__ float sK[NC * LK];
  __shared__ float sE[NC * 64];
  __shared__ float sv[NS], sqv[NF], sf[NS], sk[NC], st1[NS], sx[NF];

  const int lane = threadIdx.x;
  const int b = blockIdx.x;
  const int mn = lane & 15;
  const int h = lane >> 4;

  // ---- Backward pass: t = T-1 terminal factor ----
  {
    const size_t tb = (size_t)(TT - 1) * BATCH + b;
    g2s(sQ, LQ, Cg + tb * (size_t)(NF * NF), NF, NF, lane);
    for (int c2 = lane; c2 < NF; c2 += 32) sqv[c2] = cg[tb * (size_t)NF + c2];
    __syncthreads();
    factor_update(sQ, sqv, sE, sK, sk, sU, sS, sZ, sV, sv, st1, Kws, kws, tb, lane, h);
  }

  // ---- Backward Riccati recursion t = T-2 .. 0 ----
  for (int t = TT - 2; t >= 0; --t) {
    const size_t tb = (size_t)t * BATCH + b;
    g2s(sF, LF, Fg + tb * (size_t)(NS * NF), NS, NF, lane);
    sf[lane] = fg[tb * (size_t)NS + lane];
    g2s(sQ, LQ, Cg + tb * (size_t)(NF * NF), NF, NF, lane);
    for (int c2 = lane; c2 < NF; c2 += 32) sqv[c2] = cg[tb * (size_t)NF + c2];
    __syncthreads();

    // Prefetch next iteration's C and F tiles (global_prefetch_b8)
    if (t > 0) {
      const float* nC = Cg + ((size_t)(t - 1) * BATCH + b) * (size_t)(NF * NF);
      for (int i = lane * 32; i < NF * NF; i += 1024) __builtin_prefetch(nC + i, 0, 0);
      const float* nF = Fg + ((size_t)(t - 1) * BATCH + b) * (size_t)(NS * NF);
      for (int i = lane * 32; i < NS * NF; i += 1024) __builtin_prefetch(nF + i, 0, 0);
    }

    // W = V @ F  (32x48)
#pragma unroll
    for (int ti = 0; ti < 2; ++ti)
#pragma unroll
      for (int tj = 0; tj < 3; ++tj) {
        v8f acc = zero8();
        acc = mma_k16<false>(&sV[(16 * ti) * LV + 0], LV, &sF[0 * LF + 16 * tj], LF, acc, mn, h);
        acc = mma_k16<false>(&sV[(16 * ti) * LV + 16], LV, &sF[16 * LF + 16 * tj], LF, acc, mn, h);
        acc_store(&sW[(16 * ti) * LW + 16 * tj], LW, acc, mn, h);
      }
    __syncthreads();

    // Qm = C + F^T @ W  (48x48), accumulator seeded with C tile
#pragma unroll
    for (int ti = 0; ti < 3; ++ti)
#pragma unroll
      for (int tj = 0; tj < 3; ++tj) {
        v8f acc = acc_load(&sQ[(16 * ti) * LQ + 16 * tj], LQ, mn, h);
        acc = mma_k16<true>(&sF[0 * LF + 16 * ti], LF, &sW[0 * LW + 16 * tj], LW, acc, mn, h);
        acc = mma_k16<true>(&sF[16 * LF + 16 * ti], LF, &sW[16 * LW + 16 * tj], LW, acc, mn, h);
        acc_store(&sQ[(16 * ti) * LQ + 16 * tj], LQ, acc, mn, h);
      }
    __syncthreads();

    // symmetrize Qm
    for (int i = 0; i < NF; ++i)
      for (int j2 = lane; j2 < NF; j2 += 32)
        if (j2 > i) {
          const float a2 = 0.5f * (sQ[i * LQ + j2] + sQ[j2 * LQ + i]);
          sQ[i * LQ + j2] = a2;
          sQ[j2 * LQ + i] = a2;
        }
    __syncthreads();

    // qv = c + F^T (v + V f)
    {
      float s = sv[lane];
      for (int j2 = 0; j2 < NS; ++j2) s += sV[lane * LV + j2] * sf[j2];
      st1[lane] = s;
    }
    __syncthreads();
    for (int j2 = lane; j2 < NF; j2 += 32) {
      float s = sqv[j2];
      for (int s2 = 0; s2 < NS; ++s2) s += sF[s2 * LF + j2] * st1[s2];
      sqv[j2] = s;
    }
    __syncthreads();

    factor_update(sQ, sqv, sE, sK, sk, sU, sS, sZ, sV, sv, st1, Kws, kws, tb, lane, h);
  }

  // ---- Forward rollout ----
  sx[lane] = x0g[(size_t)b * NS + lane];
  __syncthreads();
  for (int t = 0; t < TT; ++t) {
    const size_t tb = (size_t)t * BATCH + b;
    for (int r = 0; r < 16; ++r)
      sK[r * LK + lane] = Kws[tb * (size_t)(NC * NS) + r * NS + lane];
    if (lane < 16) sk[lane] = kws[tb * (size_t)NC + lane];
    __syncthreads();
    if (lane < 16) {
      float s = sk[lane];
      for (int j2 = 0; j2 < NS; ++j2) s += sK[lane * LK + j2] * sx[j2];
      sx[NS + lane] = s;  // u
    }
    __syncthreads();
    {
      float* o = out + tb * (size_t)NF;
      for (int c2 = lane; c2 < NF; c2 += 32) o[c2] = sx[c2];
    }
    if (t < TT - 1) {
      g2s(sF, LF, Fg + tb * (size_t)(NS * NF), NS, NF, lane);
      __syncthreads();
      float s = fg[tb * (size_t)NS + lane];
      for (int j2 = 0; j2 < NF; ++j2) s += sF[lane * LF + j2] * sx[j2];
      st1[lane] = s;
      __syncthreads();
      sx[lane] = st1[lane];
      __syncthreads();
    }
  }
}

extern "C" void kernel_launch(void* const* d_in, const int* in_sizes, int n_in,
                              void* d_out, int out_size, void* d_ws, size_t ws_size,
                              hipStream_t stream) {
  const float* x0 = (const float*)d_in[0];
  const float* C  = (const float*)d_in[1];
  const float* c  = (const float*)d_in[2];
  const float* F  = (const float*)d_in[3];
  const float* f  = (const float*)d_in[4];
  float* out = (float*)d_out;
  // workspace: K (T,B,16,32) then k (T,B,16)  -> ~216 MB
  float* Kws = (float*)d_ws;
  float* kws = Kws + (size_t)TT * BATCH * (NC * NS);
  lqr_kernel<<<BATCH, 32, 0, stream>>>(x0, C, c, F, f, out, Kws, kws);
}